// MicroVideoRec_25486335935153
// MI455X (gfx1250) — compile-verified
//
#include <hip/hip_runtime.h>
#include <math.h>

#define THREADS 256

typedef int   v4i __attribute__((ext_vector_type(4)));
typedef float v4f __attribute__((ext_vector_type(4)));

// ---- order-preserving float <-> uint encoding so float max/min become
// ---- single GLOBAL_ATOMIC_MAX_U32 / MIN_U32 ops (no CAS loop) ----
__device__ __forceinline__ unsigned f32_ord(float f) {
    unsigned u = __float_as_uint(f);
    return (u & 0x80000000u) ? ~u : (u | 0x80000000u);
}
__device__ __forceinline__ float f32_unord(unsigned e) {
    return (e & 0x80000000u) ? __uint_as_float(e ^ 0x80000000u)
                             : __uint_as_float(~e);
}

// Kernel 1: reset bin state each launch (d_ws is poisoned, never re-zeroed).
__global__ void mvr_init_bins(unsigned* __restrict__ cnt, float* __restrict__ sig,
                              float* __restrict__ rep, unsigned* __restrict__ mx,
                              unsigned* __restrict__ mn, double* __restrict__ acc, int M) {
    int i = blockIdx.x * blockDim.x + threadIdx.x;
    if (i < M) {
        cnt[i] = 0u;
        sig[i] = 0.0f;
        rep[i] = 0.0f;
        mx[i]  = 0u;           // < f32_ord(x) for any real float
        mn[i]  = 0xFFFFFFFFu;  // > f32_ord(x) for any real float
    }
    if (i == 0) { acc[0] = 0.0; acc[1] = 0.0; }
}

// Kernel 2: scatter. 4 interactions per thread via 128-bit NT loads; five
// hardware atomics per interaction, all resolving in the 192MB L2
// (bin state = 20MB, fully resident).
__global__ void mvr_scatter4(const int* __restrict__ ids, const float* __restrict__ s,
                             const float* __restrict__ r, unsigned* __restrict__ cnt,
                             float* __restrict__ sig, float* __restrict__ rep,
                             unsigned* __restrict__ mx, unsigned* __restrict__ mn,
                             int n4, int N) {
    int t = blockIdx.x * blockDim.x + threadIdx.x;
    if (t >= n4) return;
    int base = t * 4;

    int   idv[4];
    float sv[4], rv[4];
    if (base + 3 < N) {
        v4i id4 = __builtin_nontemporal_load(((const v4i*)ids) + t);
        v4f s4  = __builtin_nontemporal_load(((const v4f*)s) + t);
        v4f r4  = __builtin_nontemporal_load(((const v4f*)r) + t);
        idv[0] = id4.x; idv[1] = id4.y; idv[2] = id4.z; idv[3] = id4.w;
        sv[0] = s4.x; sv[1] = s4.y; sv[2] = s4.z; sv[3] = s4.w;
        rv[0] = r4.x; rv[1] = r4.y; rv[2] = r4.z; rv[3] = r4.w;
    } else {
        #pragma unroll
        for (int k = 0; k < 4; ++k) {
            int j = base + k;
            idv[k] = (j < N) ? ids[j] : 0;
            sv[k]  = (j < N) ? s[j]   : 0.0f;
            rv[k]  = (j < N) ? r[j]   : 0.0f;
        }
    }

    #pragma unroll
    for (int k = 0; k < 4; ++k) {
        int j = base + k;
        if (j < N) {
            int id = idv[k];
            atomicAdd(&cnt[id], 1u);              // global_atomic_add_u32
            unsafeAtomicAdd(&sig[id], sv[k]);     // global_atomic_add_f32
            unsafeAtomicAdd(&rep[id], rv[k]);     // global_atomic_add_f32
            unsigned e = f32_ord(sv[k]);
            atomicMax(&mx[id], e);                // global_atomic_max_u32
            atomicMin(&mn[id], e);                // global_atomic_min_u32
        }
    }
}

// Kernel 3: per-item finalize + f64 block reduction of (rep_log, rep_log^2)
// via wave32 shuffles + LDS, one pair of global_atomic_add_f64 per block.
__global__ void mvr_finalize(const unsigned* __restrict__ cnt, const float* __restrict__ sig,
                             const float* __restrict__ rep, const unsigned* __restrict__ mx,
                             const unsigned* __restrict__ mn, const float* __restrict__ lam_raw,
                             float* __restrict__ out, double* __restrict__ acc, int M) {
    int i = blockIdx.x * blockDim.x + threadIdx.x;
    float replog = 0.0f;
    if (i < M) {
        unsigned c = cnt[i];
        float out0 = 0.0f;
        if (c) {
            float fc    = (float)c;
            float smean = sig[i] / fc;
            float repf  = rep[i] / fc;
            float mp    = f32_unord(mx[i]);
            float mneg  = f32_unord(mn[i]);
            float maxab = (fabsf(mp) >= fabsf(mneg)) ? mp : mneg;
            float lam   = 1.0f / (1.0f + expf(-lam_raw[0]));
            out0   = smean + lam * maxab;
            replog = log1pf(repf);
        }
        out[i]     = out0;    // row 0: signal_full
        out[M + i] = replog;  // row 1: rep_log (normalized by kernel 4)
    }

    // block reduction in f64 for numerically safe variance over 1M items
    double v  = (double)replog;
    double v2 = v * v;
    #pragma unroll
    for (int off = 16; off > 0; off >>= 1) {   // wave32 tree
        v  += __shfl_down(v,  off, 32);
        v2 += __shfl_down(v2, off, 32);
    }
    __shared__ double ls[THREADS / 32];
    __shared__ double ls2[THREADS / 32];
    int lane = threadIdx.x & 31;
    int wid  = threadIdx.x >> 5;
    if (lane == 0) { ls[wid] = v; ls2[wid] = v2; }
    __syncthreads();
    if (wid == 0) {
        v  = (lane < THREADS / 32) ? ls[lane]  : 0.0;
        v2 = (lane < THREADS / 32) ? ls2[lane] : 0.0;
        #pragma unroll
        for (int off = (THREADS / 32) / 2; off > 0; off >>= 1) {
            v  += __shfl_down(v,  off, 32);
            v2 += __shfl_down(v2, off, 32);
        }
        if (lane == 0) {
            unsafeAtomicAdd(&acc[0], v);   // global_atomic_add_f64
            unsafeAtomicAdd(&acc[1], v2);
        }
    }
}

// Kernel 4: normalize row 1 with mean/std (ddof=1) + 1e-6.
__global__ void mvr_normalize(float* __restrict__ out, const double* __restrict__ acc, int M) {
    int i = blockIdx.x * blockDim.x + threadIdx.x;
    if (i >= M) return;
    double sum   = acc[0];
    double sumsq = acc[1];
    double dm    = (double)M;
    double mean  = sum / dm;
    double var   = (sumsq - sum * sum / dm) / (dm - 1.0);
    var = var > 0.0 ? var : 0.0;
    float stdv = (float)sqrt(var) + 1e-6f;
    float x = out[M + i];
    out[M + i] = (x - (float)mean) / stdv;
}

extern "C" void kernel_launch(void* const* d_in, const int* in_sizes, int n_in,
                              void* d_out, int out_size, void* d_ws, size_t ws_size,
                              hipStream_t stream) {
    const int*   ids     = (const int*)d_in[0];
    const float* sig_in  = (const float*)d_in[1];
    const float* rep_in  = (const float*)d_in[2];
    const float* lam_raw = (const float*)d_in[3];

    const int N = in_sizes[0];
    const int M = out_size / 2;

    char* ws = (char*)d_ws;
    unsigned* cnt = (unsigned*)(ws);
    float*    sig = (float*)   (ws + (size_t)M * 4);
    float*    rep = (float*)   (ws + (size_t)M * 8);
    unsigned* mx  = (unsigned*)(ws + (size_t)M * 12);
    unsigned* mn  = (unsigned*)(ws + (size_t)M * 16);
    size_t accOff = (((size_t)M * 20) + 15) & ~(size_t)15;
    double*   acc = (double*)(ws + accOff);

    float* out = (float*)d_out;

    const int gbM = (M + THREADS - 1) / THREADS;
    const int n4  = (N + 3) / 4;
    const int gbN = (n4 + THREADS - 1) / THREADS;

    mvr_init_bins<<<gbM, THREADS, 0, stream>>>(cnt, sig, rep, mx, mn, acc, M);
    mvr_scatter4 <<<gbN, THREADS, 0, stream>>>(ids, sig_in, rep_in, cnt, sig, rep, mx, mn, n4, N);
    mvr_finalize <<<gbM, THREADS, 0, stream>>>(cnt, sig, rep, mx, mn, lam_raw, out, acc, M);
    mvr_normalize<<<gbM, THREADS, 0, stream>>>(out, acc, M);
}